// VectorizedQuantumLayer_21397527069310
// MI455X (gfx1250) — compile-verified
//
#include <hip/hip_runtime.h>

#define NQ 6
#define DIM 64
#define BLOCK 128
#define LROW 66   // LDS row stride in dwords: 66 mod 64 = 2 -> conflict-free columns

typedef __attribute__((ext_vector_type(2))) float v2f;
typedef __attribute__((ext_vector_type(8))) float v8f;

__device__ __forceinline__ float zsign_val(int k, int n) {
    // zsign[k][j] = 1 - 2*((k >> (5-j)) & 1); pad columns j>=6 with 0
    if (n >= NQ) return 0.0f;
    return ((k >> (5 - n)) & 1) ? -1.0f : 1.0f;
}

__global__ __launch_bounds__(BLOCK) void qsim_kernel(
    const float* __restrict__ x, const float* __restrict__ w,
    float* __restrict__ out, int B)
{
    __shared__ float ldsP[BLOCK * LROW];   // 33,792 bytes

    const int tidg = blockIdx.x * BLOCK + threadIdx.x;
    const int tid  = tidg < B ? tidg : B - 1;   // clamp: keep EXEC full for WMMA

    // Per-qubit RY(pi*tanh(x_i)) half-angle cos/sin (same angle every layer)
    float rc[NQ], rs[NQ];
    #pragma unroll
    for (int i = 0; i < NQ; ++i) {
        float xi = x[(size_t)tid * NQ + i];
        float th = 1.5707963267948966f * tanhf(xi);  // 0.5 * pi * tanh(x)
        float s, c;
        sincosf(th, &s, &c);
        rc[i] = c; rs[i] = s;
    }

    // State vector fully register-resident: 64 complex amplitudes
    float sr[DIM], si[DIM];
    #pragma unroll
    for (int k = 0; k < DIM; ++k) { sr[k] = (k == 0) ? 1.0f : 0.0f; si[k] = 0.0f; }

    #pragma unroll 1
    for (int layer = 0; layer < 3; ++layer) {
        #pragma unroll
        for (int i = 0; i < NQ; ++i) {
            // uniform scalar loads of weights
            const float* wp = w + ((layer * NQ) + i) * 3;
            float w0 = wp[0], w1 = wp[1], w2 = wp[2];
            // M = RZ(w2) RY(w1) RZ(w0) in closed form:
            // M00 =  cb*e^{-i(w2+w0)/2}   M01 = -sb*e^{-i(w2-w0)/2}
            // M10 =  sb*e^{+i(w2-w0)/2}   M11 =  cb*e^{+i(w2+w0)/2}
            float ap = 0.5f * (w2 + w0), am = 0.5f * (w2 - w0), hb = 0.5f * w1;
            float cp, sp_, cm, sm, cb, sb;
            sincosf(ap, &sp_, &cp);
            sincosf(am, &sm, &cm);
            sincosf(hb, &sb, &cb);
            float M00r =  cb * cp, M00i = -cb * sp_;
            float M01r = -sb * cm, M01i =  sb * sm;
            float M10r =  sb * cm, M10i =  sb * sm;
            float M11r =  cb * cp, M11i =  cb * sp_;
            // g = M @ RY(theta_i)  (RY real: [[c,-s],[s,c]])
            float c = rc[i], s = rs[i];
            float g00r = M00r*c + M01r*s, g00i = M00i*c + M01i*s;
            float g01r = M01r*c - M00r*s, g01i = M01i*c - M00i*s;
            float g10r = M10r*c + M11r*s, g10i = M10i*c + M11i*s;
            float g11r = M11r*c - M10r*s, g11i = M11i*c - M10i*s;

            const int mask = 1 << (5 - i);   // wire i -> bit (5-i), wire 0 = MSB
            #pragma unroll
            for (int k0 = 0; k0 < DIM; ++k0) {
                if (k0 & mask) continue;
                const int k1 = k0 | mask;
                float ar0 = sr[k0], ai0 = si[k0];
                float ar1 = sr[k1], ai1 = si[k1];
                sr[k0] = g00r*ar0 - g00i*ai0 + g01r*ar1 - g01i*ai1;
                si[k0] = g00r*ai0 + g00i*ar0 + g01r*ai1 + g01i*ar1;
                sr[k1] = g10r*ar0 - g10i*ai0 + g11r*ar1 - g11i*ai1;
                si[k1] = g10r*ai0 + g10i*ar0 + g11r*ai1 + g11i*ar1;
            }
        }
        // CNOT ring: (0,1)..(4,5),(5,0) -> compile-time register swaps
        #pragma unroll
        for (int cc = 0; cc < NQ; ++cc) {
            const int ctrl = cc, tgt = (cc + 1) % NQ;
            const int pc = 5 - ctrl, pt = 5 - tgt;
            #pragma unroll
            for (int k = 0; k < DIM; ++k) {
                if (((k >> pc) & 1) && !((k >> pt) & 1)) {
                    const int k2 = k | (1 << pt);
                    float t0 = sr[k]; sr[k] = sr[k2]; sr[k2] = t0;
                    float t1 = si[k]; si[k] = si[k2]; si[k2] = t1;
                }
            }
        }
    }

    // Stage probabilities into LDS (row = thread, skewed stride -> no bank conflicts)
    const int row = threadIdx.x;
    #pragma unroll
    for (int k = 0; k < DIM; ++k)
        ldsP[row * LROW + k] = sr[k]*sr[k] + si[k]*si[k];

    __syncthreads();

    // Readout GEMM: probs[16x64] @ zsign[64x16] via V_WMMA_F32_16X16X4_F32
    // A layout (32-bit A 16x4): lane 0-15 -> M, VGPR0=K0 / VGPR1=K1; lanes 16-31 same M, K+2/K+3
    // B layout assumed mirror: lane%16 -> N, VGPR0=K(0|2), VGPR1=K(1|3)
    const int wv    = threadIdx.x >> 5;
    const int lane  = threadIdx.x & 31;
    const int laneM = lane & 15;
    const int hi    = lane >> 4;

    #pragma unroll
    for (int t = 0; t < 2; ++t) {
        const int rowBase = wv * 32 + t * 16;
        const int r = rowBase + laneM;
        v8f acc = {0.f, 0.f, 0.f, 0.f, 0.f, 0.f, 0.f, 0.f};
        #pragma unroll
        for (int cch = 0; cch < 16; ++cch) {
            const int kb = 4 * cch + 2 * hi;
            v2f a, b;
            a.x = ldsP[r * LROW + kb];
            a.y = ldsP[r * LROW + kb + 1];
            b.x = zsign_val(kb,     laneM);
            b.y = zsign_val(kb + 1, laneM);
            // (neg_a, A, neg_b, B, c_mod, C, reuse_a, reuse_b)
            acc = __builtin_amdgcn_wmma_f32_16x16x4_f32(
                false, a, false, b, (short)0, acc, false, false);
        }
        // D layout: VGPR vd -> row M = vd + 8*hi, lane%16 -> N
        if (laneM < NQ) {
            const int sampleBase = blockIdx.x * BLOCK + rowBase;
            #pragma unroll
            for (int vd = 0; vd < 8; ++vd) {
                const int smp = sampleBase + vd + 8 * hi;
                if (smp < B) out[(size_t)smp * NQ + laneM] = acc[vd];
            }
        }
    }
}

extern "C" void kernel_launch(void* const* d_in, const int* in_sizes, int n_in,
                              void* d_out, int out_size, void* d_ws, size_t ws_size,
                              hipStream_t stream) {
    (void)n_in; (void)out_size; (void)d_ws; (void)ws_size;
    const float* x = (const float*)d_in[0];   // [B, 6] f32
    const float* w = (const float*)d_in[1];   // [3, 6, 3] f32
    float* out = (float*)d_out;               // [B, 6] f32
    const int B = in_sizes[0] / NQ;
    const int grid = (B + BLOCK - 1) / BLOCK;
    qsim_kernel<<<grid, BLOCK, 0, stream>>>(x, w, out, B);
}